// SpatialCrossAttention_19361712570500
// MI455X (gfx1250) — compile-verified
//
#include <hip/hip_runtime.h>
#include <hip/hip_bf16.h>

// ---------------- problem constants (static in the reference) ----------------
#define CAMS   6
#define BS     2
#define BC     (CAMS*BS)     // 12
#define NQ     2500
#define HEADS  8
#define LVLS   4
#define PP     2
#define ZZ     4
#define PZ     (PP*ZZ)       // 8
#define NK     32            // L*P*Z
#define DIM    256
#define DH     (DIM/HEADS)   // 32
#define MQ     (BS*NQ)       // 5000
#define NOFF   (HEADS*LVLS*PZ*2) // 512

#define KC        128        // K-chunk staged in LDS per TDM transfer
#define LDSROW    132        // dwords per staged row: 128 + 4 pad  (132 mod 64 == 4
                             //  -> the wave's 32 ds_load_b64 hit all 64 banks once)

typedef float v2f __attribute__((ext_vector_type(2)));
typedef float v8f __attribute__((ext_vector_type(8)));
typedef unsigned int v4u __attribute__((ext_vector_type(4)));
typedef int v4i __attribute__((ext_vector_type(4)));
typedef int v8i __attribute__((ext_vector_type(8)));

// spatial_shapes = [[4,4],[4,2],[2,2],[2,2]]  (H,W); starts into nk
__constant__ int   c_H[LVLS]    = {4, 4, 2, 2};
__constant__ int   c_W[LVLS]    = {4, 2, 2, 2};
__constant__ int   c_start[LVLS]= {0, 16, 24, 28};
__constant__ float c_invW[LVLS] = {0.25f, 0.5f, 0.5f, 0.5f};
__constant__ float c_invH[LVLS] = {0.25f, 0.25f, 0.5f, 0.5f};

// =====================================================================
// TDM 2D tile load: global(row-major, stride tensor_stride elems, fp32)
//   -> LDS, tile = tile_rows x KC elems, LDS row stride = LDSROW dwords
// (pad_enable: every 128 dwords insert 4 dwords). Rows beyond valid_rows
// are zero-filled by TDM OOB handling (tensor_dim1 = valid_rows).
// Descriptor packing per CDNA5 ISA.
// This toolchain exposes the 6-arg builtin: (g0, g1, g2, g3, g4, cpol).
// =====================================================================
__device__ __forceinline__
void tdm_load_tile(void* lds_dst, const void* gsrc,
                   int tile_rows, int valid_rows, int tensor_stride) {
    const unsigned long long ga = (unsigned long long)gsrc;
    const unsigned lds_addr = (unsigned)(unsigned long long)lds_dst; // low 32b = LDS offset
    // group0: count=1 | lds_addr | global_addr[56:0] | type=2
    v4u g0 = { 1u,
               lds_addr,
               (unsigned)(ga & 0xFFFFFFFFull),
               (unsigned)((ga >> 32) & 0x01FFFFFFull) | (2u << 30) };
    // group1: data_size=2(4B), pad_enable, pad_interval=6(128dw), pad_amount=3(4dw)
    //   tensor_dim0=KC (chunk fully in-bounds in X), tensor_dim1=valid_rows,
    //   tile_dim0=KC, tile_dim1=tile_rows, tile_dim2=0,
    //   tensor_dim0_stride=tensor_stride, tensor_dim1_stride unused.
    v8i g1 = { (int)((2u << 16) | (1u << 20) | (6u << 22) | (3u << 25)),
               (int)((unsigned)KC << 16),                  // tensor_dim0[15:0] @ bits[63:48]
               (int)((unsigned)(valid_rows & 0xFFFF) << 16), // tensor_dim1[15:0] @ bits[95:80]
               (int)((unsigned)KC << 16),                  // tile_dim0 @ bits[127:112]
               (int)(tile_rows & 0xFFFF),                  // tile_dim1 @ bits[143:128]
               tensor_stride,                              // tensor_dim0_stride[31:0]
               0, 0 };
    v4i z4 = {0, 0, 0, 0};
    v8i z8 = {0, 0, 0, 0, 0, 0, 0, 0};
    __builtin_amdgcn_tensor_load_to_lds(g0, g1, z4, z4, z8, 0);
}

// =====================================================================
// Fp32 WMMA GEMM:  C[M,N] = A[M,K] @ W[N,K]^T  (+bias[N]) (+R[M,N])
// block = 128 threads (4 waves). Block tile = 16(M) x 64(N).
// A/W tiles staged in LDS by the Tensor Data Mover in KC-deep chunks;
// WMMA operands come from conflict-free ds_load_b64.
// grid.x = ceil(M/16), grid.y = N/64. K, N multiples of KC/64 resp.
// =====================================================================
__global__ void __launch_bounds__(128)
wmma_gemm_xwt(const float* __restrict__ A,
              const float* __restrict__ W,
              const float* __restrict__ bias,
              const float* __restrict__ R,
              float* __restrict__ C,
              int M, int N, int K) {
    __shared__ float sA[16 * LDSROW];
    __shared__ float sW[64 * LDSROW];

    const int wave = threadIdx.x >> 5;
    const int lane = threadIdx.x & 31;
    const int m0   = blockIdx.x * 16;
    const int nblk = blockIdx.y * 64;
    const int n0   = nblk + wave * 16;

    const int hi  = lane >> 4;              // 0 | 1
    const int idx = lane & 15;
    const int kb  = hi * 2;

    int rowsA = M - m0; if (rowsA > 16) rowsA = 16;   // TDM zero-fills the rest

    v8f acc = {};
    for (int kc = 0; kc < K; kc += KC) {
        __syncthreads();                    // previous chunk fully consumed
        if (wave == 0) {
            tdm_load_tile(sA, A + (size_t)m0 * K + kc, 16, rowsA, K);
            tdm_load_tile(sW, W + (size_t)nblk * K + kc, 64, 64, K);
            __builtin_amdgcn_s_wait_tensorcnt(0);
        }
        __syncthreads();                    // LDS tiles visible to all waves

        #pragma unroll 8
        for (int k0 = 0; k0 < KC; k0 += 4) {
            // fp32 16x16x4 WMMA operand layout: lane(hi,idx) supplies
            //   A[m=idx][k+2*hi .. +1] and B[k+2*hi .. +1][n=idx]
            v2f a = *(const v2f*)(&sA[idx * LDSROW + k0 + kb]);
            v2f b = *(const v2f*)(&sW[(wave * 16 + idx) * LDSROW + k0 + kb]);
            acc = __builtin_amdgcn_wmma_f32_16x16x4_f32(
                      false, a, false, b, (short)0, acc, false, false);
        }
    }

    #pragma unroll
    for (int r = 0; r < 8; ++r) {
        const int m = m0 + r + 8 * hi;      // D layout: VGPR r -> M = r / 8+r
        const int n = n0 + idx;
        if (m < M) {
            float v = acc[r];
            if (bias) v += bias[n];
            if (R)    v += R[(size_t)m * N + n];
            C[(size_t)m * N + n] = v;
        }
    }
}

// =====================================================================
// Fused attention + multi-level bilinear sampling.
// One wave per (rebatch j, head h, query qi).
//   phase 1: lane = key index (32 keys) -> logits, softmax via shfl_xor
//   phase 2: lane = channel d (32 dims) -> weighted grid-sample accumulate
// q-side decoded bs-major (b=j/CAMS, cam=j%CAMS); k/v indexed flat by j
// (faithful to the reference's cam-major key_rb quirk).
// =====================================================================
__global__ void attn_sample_kernel(const float* __restrict__ q_buf,   // [BS,NQ,DIM]
                                   const float* __restrict__ k_buf,   // [BC,NK,DIM]
                                   const float* __restrict__ v_buf,   // [BC,NK,DIM]
                                   const float* __restrict__ off_buf, // [BS,NQ,512]
                                   const float* __restrict__ rpc,     // [CAMS,BS,NQ,ZZ,2]
                                   float* __restrict__ att_buf) {     // [BC,NQ,DIM]
    const int lane = threadIdx.x & 31;
    const int wv   = blockIdx.x * (blockDim.x >> 5) + (threadIdx.x >> 5);
    if (wv >= BC * HEADS * NQ) return;

    const int j   = wv / (HEADS * NQ);
    const int rem = wv % (HEADS * NQ);
    const int h   = rem / NQ;
    const int qi  = rem % NQ;
    const int b   = j / CAMS;
    const int cam = j % CAMS;

    // ---- logits: lane owns key kk = lane ----
    const float* __restrict__ qv = q_buf + ((size_t)b * NQ + qi) * DIM + h * DH;
    const float* __restrict__ kv = k_buf + ((size_t)j * NK + lane) * DIM + h * DH;
    float lg = 0.0f;
    #pragma unroll
    for (int dd = 0; dd < DH; ++dd) lg = fmaf(qv[dd], kv[dd], lg);

    // ---- softmax across the 32 lanes ----
    float mx = lg;
    for (int o = 16; o > 0; o >>= 1) mx = fmaxf(mx, __shfl_xor(mx, o, 32));
    float ex = __expf(lg - mx);
    float sm = ex;
    for (int o = 16; o > 0; o >>= 1) sm += __shfl_xor(sm, o, 32);
    const float attn = ex / sm;              // lane holds attn[key=lane]

    // ---- sampling: lane owns channel dd = lane ----
    const float* __restrict__ offp = off_buf + ((size_t)b * NQ + qi) * NOFF;
    const float* __restrict__ refp = rpc + (((size_t)cam * BS + b) * NQ + qi) * (ZZ * 2);
    const float* __restrict__ vlane = v_buf + (size_t)j * NK * DIM + h * DH + lane;

    float acc = 0.0f;
    #pragma unroll
    for (int kk = 0; kk < NK; ++kk) {
        const int l  = kk >> 3;
        const int pz = kk & 7;
        const int z  = pz & 3;
        const float a_k = __shfl(attn, kk, 32);

        const float ox = offp[((h * LVLS + l) * PZ + pz) * 2 + 0] * c_invW[l];
        const float oy = offp[((h * LVLS + l) * PZ + pz) * 2 + 1] * c_invH[l];
        const float gx = 2.0f * (refp[z * 2 + 0] + ox) - 1.0f;
        const float gy = 2.0f * (refp[z * 2 + 1] + oy) - 1.0f;

        const int W_ = c_W[l], H_ = c_H[l];
        const float x = (gx + 1.0f) * (W_ * 0.5f) - 0.5f;
        const float y = (gy + 1.0f) * (H_ * 0.5f) - 0.5f;
        const float x0f = floorf(x), y0f = floorf(y);
        const int x0 = (int)x0f, y0 = (int)y0f;
        const float wx1 = x - x0f, wy1 = y - y0f;
        const float wx0 = 1.0f - wx1, wy0 = 1.0f - wy1;

        const float* __restrict__ vb = vlane + (size_t)c_start[l] * DIM;
        auto corner = [&](int xi, int yi, float w) -> float {
            const bool ok = (xi >= 0) & (xi < W_) & (yi >= 0) & (yi < H_);
            const int xc = min(max(xi, 0), W_ - 1);
            const int yc = min(max(yi, 0), H_ - 1);
            const float val = vb[(size_t)(yc * W_ + xc) * DIM];   // coalesced over lanes
            return ok ? w * val : 0.0f;
        };
        float s = corner(x0, y0, wx0 * wy0) + corner(x0 + 1, y0, wx1 * wy0)
                + corner(x0, y0 + 1, wx0 * wy1) + corner(x0 + 1, y0 + 1, wx1 * wy1);
        acc = fmaf(a_k, s, acc);
    }
    att_buf[((size_t)j * NQ + qi) * DIM + h * DH + lane] = acc;
}

// =====================================================================
// Camera reduction with bev_mask semantics:
//   valid[cam,qi] from batch 0; count = clip(#valid cams per (b,qi), 1)
// =====================================================================
__global__ void cam_reduce_kernel(const float* __restrict__ att,      // [BC,NQ,DIM]
                                  const unsigned char* __restrict__ mask, // [CAMS,BS,NQ,ZZ]
                                  float* __restrict__ slots) {        // [BS,NQ,DIM]
    const int i = blockIdx.x * blockDim.x + threadIdx.x;
    if (i >= BS * NQ * DIM) return;
    const int dd = i % DIM;
    const int qi = (i / DIM) % NQ;
    const int b  = i / (DIM * NQ);

    float cnt = 0.0f, s = 0.0f;
    #pragma unroll
    for (int cam = 0; cam < CAMS; ++cam) {
        const size_t mb  = ((size_t)(cam * BS + b) * NQ + qi) * ZZ;
        const size_t mb0 = ((size_t)(cam * BS + 0) * NQ + qi) * ZZ;
        int mz = 0, mz0 = 0;
        #pragma unroll
        for (int z = 0; z < ZZ; ++z) { mz += mask[mb + z]; mz0 += mask[mb0 + z]; }
        if (mz  > 0) cnt += 1.0f;
        if (mz0 > 0) s += att[((size_t)(b * CAMS + cam) * NQ + qi) * DIM + dd];
    }
    slots[i] = s / fmaxf(cnt, 1.0f);
}

// =====================================================================
extern "C" void kernel_launch(void* const* d_in, const int* in_sizes, int n_in,
                              void* d_out, int out_size, void* d_ws, size_t ws_size,
                              hipStream_t stream) {
    const float* query = (const float*)d_in[0];   // [BS,NQ,DIM]
    const float* key   = (const float*)d_in[1];   // [CAMS,BS,NK,DIM] -> flat [BC*NK,DIM]
    const float* value = (const float*)d_in[2];
    const float* rpc   = (const float*)d_in[4];   // [CAMS,BS,NQ,ZZ,2]
    const unsigned char* mask = (const unsigned char*)d_in[5]; // bool [CAMS,BS,NQ,ZZ]
    const float* Wq   = (const float*)d_in[7];   const float* bq   = (const float*)d_in[8];
    const float* Wk   = (const float*)d_in[9];   const float* bk   = (const float*)d_in[10];
    const float* Wv   = (const float*)d_in[11];  const float* bv   = (const float*)d_in[12];
    const float* Woff = (const float*)d_in[13];  const float* boff = (const float*)d_in[14];
    const float* Wout = (const float*)d_in[15];  const float* bout = (const float*)d_in[16];
    float* out = (float*)d_out;

    // workspace carve-up
    char* ws = (char*)d_ws;
    float* q_buf   = (float*)ws; ws += (size_t)MQ * DIM  * sizeof(float);  // 5.1 MB
    float* off_buf = (float*)ws; ws += (size_t)MQ * NOFF * sizeof(float);  // 10.2 MB
    float* k_buf   = (float*)ws; ws += (size_t)BC * NK * DIM * sizeof(float);
    float* v_buf   = (float*)ws; ws += (size_t)BC * NK * DIM * sizeof(float);
    float* att_buf = (float*)ws; ws += (size_t)BC * NQ * DIM * sizeof(float); // 30.7 MB
    float* slots   = (float*)ws; ws += (size_t)MQ * DIM * sizeof(float);

    const dim3 gblk(128);
    const int mtq = (MQ + 15) / 16;          // 313
    const int mtk = (BC * NK + 15) / 16;     // 24

    // q = query @ Wq^T + bq            [MQ, 256]
    wmma_gemm_xwt<<<dim3(mtq, DIM / 64), gblk, 0, stream>>>(query, Wq, bq, nullptr, q_buf, MQ, DIM, DIM);
    // offsets = query @ Woff^T + boff  [MQ, 512]
    wmma_gemm_xwt<<<dim3(mtq, NOFF / 64), gblk, 0, stream>>>(query, Woff, boff, nullptr, off_buf, MQ, NOFF, DIM);
    // k = key_rb @ Wk^T + bk           [384, 256]
    wmma_gemm_xwt<<<dim3(mtk, DIM / 64), gblk, 0, stream>>>(key, Wk, bk, nullptr, k_buf, BC * NK, DIM, DIM);
    // v = value_rb @ Wv^T + bv
    wmma_gemm_xwt<<<dim3(mtk, DIM / 64), gblk, 0, stream>>>(value, Wv, bv, nullptr, v_buf, BC * NK, DIM, DIM);

    // fused softmax(qk) + multi-level bilinear sample -> att_buf
    const int total_waves = BC * HEADS * NQ;             // 240000
    const int blocks = (total_waves + 7) / 8;            // 8 waves / 256-thread block
    attn_sample_kernel<<<blocks, 256, 0, stream>>>(q_buf, k_buf, v_buf, off_buf, rpc, att_buf);

    // camera mask reduction -> slots
    cam_reduce_kernel<<<(BS * NQ * DIM + 255) / 256, 256, 0, stream>>>(att_buf, mask, slots);

    // out = slots @ Wout^T + bout + query  (residual fused into GEMM epilogue)
    wmma_gemm_xwt<<<dim3(mtq, DIM / 64), gblk, 0, stream>>>(slots, Wout, bout, query, out, MQ, DIM, DIM);
}